// DualStateReasoning_80908593923432
// MI455X (gfx1250) — compile-verified
//
#include <hip/hip_runtime.h>
#include <math.h>

// ---------------------------------------------------------------------------
// DualStateReasoning for MI455X (gfx1250, wave32, WMMA).
// Strategy: defer K/V projections through the attention (single-query decode),
// making the whole thing bandwidth-bound (~5.4 GB total -> ~230us @ 23.3TB/s).
// All GEMMs use V_WMMA_F32_16X16X4_F32 (fp32 WMMA, matches fp32 reference).
// Inner loops are branch-free (index clamping instead of load masking) and
// K-unrolled x4 so loads pipeline ahead of the WMMA chain.
// ---------------------------------------------------------------------------

typedef __attribute__((ext_vector_type(2))) float v2f;
typedef __attribute__((ext_vector_type(8))) float v8f;

#define NWAVES 8           // waves per block; K is split across waves
#define KSTEP  (4 * NWAVES)
#define UNR    4
#define BSZ    16          // batch
#define DMODEL 2048
#define NHEAD  8
#define HDIM   256

__device__ __forceinline__ float actf(float v, int act) {
  if (act == 1) return tanhf(v);
  if (act == 2) return 1.0f / (1.0f + __expf(-v));
  return v;
}

__device__ __forceinline__ int imin(int a, int b) { return a < b ? a : b; }

// ---------------------------------------------------------------------------
// GEMM-AT: out[m,n] = act( alpha * ( sum_k A[m,k]*W[n,k] + bias[n] ) )
// A rows: chunked "concat memory" addressing when chunk>0, else dense.
// Row/col indices are clamped (not masked): out-of-range lanes compute
// duplicates of the last valid row/col, which the epilogue never stores.
// ---------------------------------------------------------------------------
__global__ __launch_bounds__(32 * NWAVES)
void gemm_at_k(const float* __restrict__ A, long long ams, long long abst,
               int chunk, long long chunk_ld,
               const float* __restrict__ W, long long wns, long long wbs,
               const float* __restrict__ bias, long long bbs,
               float* __restrict__ out, long long oms, long long obs,
               int M, int N, int K, float alpha, int act)
{
  __shared__ float red[NWAVES * 32 * 8];
  const int tid  = threadIdx.x;
  const int lane = tid & 31, wave = tid >> 5;
  const int half = lane >> 4, ln15 = lane & 15;
  const int batch = blockIdx.z;
  const int m0 = blockIdx.x * 16, n0 = blockIdx.y * 16;

  const int arow = imin(m0 + ln15, M - 1);      // clamp: never OOB, dup rows unstored
  const float* aptr;
  if (chunk > 0) {
    const int ci = arow / chunk;
    const int cr = arow - ci * chunk;
    aptr = A + ((long long)(ci * BSZ + batch) * chunk + cr) * chunk_ld;
  } else {
    aptr = A + (long long)batch * abst + (long long)arow * ams;
  }
  const int nn = imin(n0 + ln15, N - 1);
  const float* wptr = W + (long long)batch * wbs + (long long)nn * wns;

  v8f c = {};
  const int kmain = K - (K % (KSTEP * UNR));
  int kk = wave * 4;
  for (; kk < kmain; kk += KSTEP * UNR) {
    float2 af[UNR], bf[UNR];
#pragma unroll
    for (int u = 0; u < UNR; u++) {
      const int k2 = kk + u * KSTEP + half * 2;
      af[u] = *(const float2*)(aptr + k2);
      bf[u] = *(const float2*)(wptr + k2);
    }
#pragma unroll
    for (int u = 0; u < UNR; u++) {
      v2f a, b;
      a.x = af[u].x; a.y = af[u].y;
      b.x = bf[u].x; b.y = bf[u].y;
      c = __builtin_amdgcn_wmma_f32_16x16x4_f32(false, a, false, b, (short)0, c,
                                                false, false);
    }
  }
  for (; kk < K; kk += KSTEP) {
    const int k2 = kk + half * 2;
    float2 ta = *(const float2*)(aptr + k2);
    float2 tb = *(const float2*)(wptr + k2);
    v2f a, b;
    a.x = ta.x; a.y = ta.y;
    b.x = tb.x; b.y = tb.y;
    c = __builtin_amdgcn_wmma_f32_16x16x4_f32(false, a, false, b, (short)0, c,
                                              false, false);
  }
#pragma unroll
  for (int r = 0; r < 8; r++) red[(wave * 32 + lane) * 8 + r] = c[r];
  __syncthreads();
  if (wave == 0) {
#pragma unroll
    for (int r = 0; r < 8; r++) {
      float v = 0.f;
      for (int w2 = 0; w2 < NWAVES; w2++) v += red[(w2 * 32 + lane) * 8 + r];
      const int gm = m0 + half * 8 + r;  // C/D layout: VGPR r -> rows r, r+8
      const int gn = n0 + ln15;
      if (gm < M && gn < N) {
        if (bias) v += bias[(long long)batch * bbs + gn];
        v *= alpha;
        out[(long long)batch * obs + (long long)gm * oms + gn] = actf(v, act);
      }
    }
  }
}

// ---------------------------------------------------------------------------
// GEMM-AN: out[m,n] = sum_k A[m,k] * B[k,n]
// chunk==0 : A k-contiguous (float2 frags), B rows via pointer stride.
// chunk>0  : A k-strided, B rows = chunked memory rows; per-row element
//            offsets precomputed into LDS once per block (no divides in loop).
// ---------------------------------------------------------------------------
__global__ __launch_bounds__(32 * NWAVES)
void gemm_an_k(const float* __restrict__ A, long long ams, long long aks,
               long long abst,
               const float* __restrict__ Bm, long long bks, long long bbs,
               int chunk, long long chunk_ld,
               float* __restrict__ out, long long oms, long long obs,
               int M, int N, int K)
{
  __shared__ float red[NWAVES * 32 * 8];
  __shared__ int boff[1536 + 2];
  const int tid  = threadIdx.x;
  const int lane = tid & 31, wave = tid >> 5;
  const int half = lane >> 4, ln15 = lane & 15;
  const int batch = blockIdx.z;
  const int m0 = blockIdx.x * 16, n0 = blockIdx.y * 16;

  const int arow = imin(m0 + ln15, M - 1);
  const float* aRow = A + (long long)batch * abst + (long long)arow * ams;
  const int nn = imin(n0 + ln15, N - 1);

  if (chunk > 0) {
    for (int l = tid; l < K; l += 32 * NWAVES) {
      const int ci = l / chunk;
      const int cr = l - ci * chunk;
      boff[l] = ((ci * BSZ + batch) * chunk + cr) * (int)chunk_ld;
    }
    __syncthreads();
  }

  v8f c = {};
  const int kmain = K - (K % (KSTEP * UNR));
  int kk = wave * 4;

  if (chunk > 0) {
    // A strided (attention probs), B rows from LDS offset table
    for (; kk < kmain; kk += KSTEP * UNR) {
      float ax[UNR], ay[UNR], bx[UNR], by[UNR];
#pragma unroll
      for (int u = 0; u < UNR; u++) {
        const int k2 = kk + u * KSTEP + half * 2;
        const int2 o = *(const int2*)(boff + k2);
        ax[u] = aRow[(long long)k2 * aks];
        ay[u] = aRow[(long long)(k2 + 1) * aks];
        bx[u] = Bm[(long long)o.x + nn];
        by[u] = Bm[(long long)o.y + nn];
      }
#pragma unroll
      for (int u = 0; u < UNR; u++) {
        v2f a, b;
        a.x = ax[u]; a.y = ay[u];
        b.x = bx[u]; b.y = by[u];
        c = __builtin_amdgcn_wmma_f32_16x16x4_f32(false, a, false, b, (short)0, c,
                                                  false, false);
      }
    }
    for (; kk < K; kk += KSTEP) {
      const int k2 = kk + half * 2;
      const int2 o = *(const int2*)(boff + k2);
      v2f a, b;
      a.x = aRow[(long long)k2 * aks];
      a.y = aRow[(long long)(k2 + 1) * aks];
      b.x = Bm[(long long)o.x + nn];
      b.y = Bm[(long long)o.y + nn];
      c = __builtin_amdgcn_wmma_f32_16x16x4_f32(false, a, false, b, (short)0, c,
                                                false, false);
    }
  } else {
    // A k-contiguous, B dense rows
    const float* bCol = Bm + (long long)batch * bbs + nn;
    for (; kk < kmain; kk += KSTEP * UNR) {
      float2 af[UNR];
      float bx[UNR], by[UNR];
#pragma unroll
      for (int u = 0; u < UNR; u++) {
        const int k2 = kk + u * KSTEP + half * 2;
        af[u] = *(const float2*)(aRow + k2);
        bx[u] = bCol[(long long)k2 * bks];
        by[u] = bCol[(long long)(k2 + 1) * bks];
      }
#pragma unroll
      for (int u = 0; u < UNR; u++) {
        v2f a, b;
        a.x = af[u].x; a.y = af[u].y;
        b.x = bx[u];   b.y = by[u];
        c = __builtin_amdgcn_wmma_f32_16x16x4_f32(false, a, false, b, (short)0, c,
                                                  false, false);
      }
    }
    for (; kk < K; kk += KSTEP) {
      const int k2 = kk + half * 2;
      float2 ta = *(const float2*)(aRow + k2);
      v2f a, b;
      a.x = ta.x; a.y = ta.y;
      b.x = bCol[(long long)k2 * bks];
      b.y = bCol[(long long)(k2 + 1) * bks];
      c = __builtin_amdgcn_wmma_f32_16x16x4_f32(false, a, false, b, (short)0, c,
                                                false, false);
    }
  }

#pragma unroll
  for (int r = 0; r < 8; r++) red[(wave * 32 + lane) * 8 + r] = c[r];
  __syncthreads();
  if (wave == 0) {
#pragma unroll
    for (int r = 0; r < 8; r++) {
      float v = 0.f;
      for (int w2 = 0; w2 < NWAVES; w2++) v += red[(w2 * 32 + lane) * 8 + r];
      const int gm = m0 + half * 8 + r;
      const int gn = n0 + ln15;
      if (gm < M && gn < N)
        out[(long long)batch * obs + (long long)gm * oms + gn] = v;
    }
  }
}

// column-mean over R rows: out[b*ors + d] = mean_r X[b,r,d]
__global__ void mean_k(const float* __restrict__ X, int R,
                       float* __restrict__ out, long long ors)
{
  const int b = blockIdx.y;
  const int d = blockIdx.x * 256 + threadIdx.x;
  const float* p = X + ((long long)b * R) * DMODEL + d;
  float s = 0.f;
  for (int r = 0; r < R; r++) s += p[(long long)r * DMODEL];
  out[(long long)b * ors + d] = s / (float)R;
}

// cK[b,h] = dot(qh[b, h*256 :], bk[h*256 :]) ; one block per b, wave per head
__global__ void ck_k(const float* __restrict__ qh, const float* __restrict__ bk,
                     float* __restrict__ cK)
{
  const int b = blockIdx.x;
  const int h = threadIdx.x >> 5, lane = threadIdx.x & 31;
  const float* q  = qh + (long long)b * DMODEL + h * HDIM;
  const float* bb = bk + h * HDIM;
  float s = 0.f;
  for (int d = lane; d < HDIM; d += 32) s += q[d] * bb[d];
  for (int o = 16; o > 0; o >>= 1) s += __shfl_down(s, o, 32);
  if (lane == 0) cK[b * NHEAD + h] = s;
}

// softmax over l of s[b, l*8 + h], with optional text padding mask (l % S)
__global__ void softmax_k(float* __restrict__ s, int L, const int* __restrict__ mask,
                          int S)
{
  __shared__ float sm[256];
  const int b = blockIdx.x >> 3, h = blockIdx.x & 7;
  float* row = s + (long long)b * L * NHEAD + h;
  const int* mrow = mask ? (mask + (long long)b * S) : nullptr;
  const int tid = threadIdx.x;

  float mx = -3.4e38f;
  for (int l = tid; l < L; l += 256) {
    float v = row[(long long)l * NHEAD];
    if (mrow && mrow[l % S] == 0) v = -1e9f;
    mx = fmaxf(mx, v);
  }
  sm[tid] = mx; __syncthreads();
  for (int o = 128; o > 0; o >>= 1) {
    if (tid < o) sm[tid] = fmaxf(sm[tid], sm[tid + o]);
    __syncthreads();
  }
  mx = sm[0]; __syncthreads();

  float sum = 0.f;
  for (int l = tid; l < L; l += 256) {
    float v = row[(long long)l * NHEAD];
    if (mrow && mrow[l % S] == 0) v = -1e9f;
    sum += __expf(v - mx);
  }
  sm[tid] = sum; __syncthreads();
  for (int o = 128; o > 0; o >>= 1) {
    if (tid < o) sm[tid] += sm[tid + o];
    __syncthreads();
  }
  const float inv = 1.0f / sm[0];

  for (int l = tid; l < L; l += 256) {
    float v = row[(long long)l * NHEAD];
    if (mrow && mrow[l % S] == 0) v = -1e9f;
    row[(long long)l * NHEAD] = __expf(v - mx) * inv;
  }
}

// cat = [c0 | c1 | c2 | c3], each chunk (16,2048), out row width ow
__global__ void concat4_k(const float* __restrict__ c0, const float* __restrict__ c1,
                          const float* __restrict__ c2, const float* __restrict__ c3,
                          float* __restrict__ out, long long ow)
{
  const int idx = blockIdx.x * 256 + threadIdx.x;
  const int b = idx >> 11, d = idx & 2047;
  float* o = out + (long long)b * ow + d;
  o[0 * DMODEL] = c0[idx];
  o[1 * DMODEL] = c1[idx];
  o[2 * DMODEL] = c2[idx];
  if (c3) o[3 * DMODEL] = c3[idx];
}

// cat[:, 0:2048] = r * S
__global__ void mulin_k(const float* __restrict__ r, const float* __restrict__ S,
                        float* __restrict__ out, long long ow)
{
  const int idx = blockIdx.x * 256 + threadIdx.x;
  const int b = idx >> 11, d = idx & 2047;
  out[(long long)b * ow + d] = r[idx] * S[idx];
}

// S = (1-z)*S + z*h
__global__ void gru_k(const float* __restrict__ z, const float* __restrict__ h,
                      float* __restrict__ S)
{
  const int idx = blockIdx.x * 256 + threadIdx.x;
  const float zz = z[idx];
  S[idx] = (1.0f - zz) * S[idx] + zz * h[idx];
}

__global__ void final_k(const float* __restrict__ Sv, const float* __restrict__ St,
                        float* __restrict__ out)
{
  const int idx = blockIdx.x * 256 + threadIdx.x;
  const int b = idx >> 11, d = idx & 2047;
  out[(long long)b * 4096 + d]        = Sv[idx];
  out[(long long)b * 4096 + 2048 + d] = St[idx];
}

// ------------------------------- host side ---------------------------------

static inline void la_at(hipStream_t st, const float* A, long long ams, long long abst,
                         int chunk, long long cld,
                         const float* W, long long wns, long long wbs,
                         const float* bias, long long bbs,
                         float* out, long long oms, long long obs,
                         int M, int N, int K, float alpha, int act, int nb)
{
  dim3 g((M + 15) / 16, (N + 15) / 16, nb);
  gemm_at_k<<<g, dim3(32 * NWAVES), 0, st>>>(A, ams, abst, chunk, cld, W, wns, wbs,
                                             bias, bbs, out, oms, obs, M, N, K,
                                             alpha, act);
}

static inline void la_an(hipStream_t st, const float* A, long long ams, long long aks,
                         long long abst,
                         const float* Bm, long long bks, long long bbs,
                         int chunk, long long cld,
                         float* out, long long oms, long long obs,
                         int M, int N, int K, int nb)
{
  dim3 g((M + 15) / 16, (N + 15) / 16, nb);
  gemm_an_k<<<g, dim3(32 * NWAVES), 0, st>>>(A, ams, aks, abst, Bm, bks, bbs, chunk,
                                             cld, out, oms, obs, M, N, K);
}

extern "C" void kernel_launch(void* const* d_in, const int* in_sizes, int n_in,
                              void* d_out, int out_size, void* d_ws, size_t ws_size,
                              hipStream_t stream)
{
  (void)in_sizes; (void)n_in; (void)out_size; (void)ws_size;
  const float* Mv   = (const float*)d_in[0];   // (4,16,196,2048)
  const float* Mt   = (const float*)d_in[1];   // (4,16,384,2048)
  const float* Fa   = (const float*)d_in[2];   // (16,11,2048)
  const int*   tmk  = (const int*)d_in[3];     // (16,384)
  const float* Wqkv = (const float*)d_in[4];   // (4,6144,2048)
  const float* bqkv = (const float*)d_in[5];   // (4,6144)
  const float* Wo   = (const float*)d_in[6];   // (4,2048,2048)
  const float* bo   = (const float*)d_in[7];   // (4,2048)
  const float* phiw = (const float*)d_in[8];   // (2,2048,4096)
  const float* phib = (const float*)d_in[9];   // (2,2048)
  const float* Wa   = (const float*)d_in[10];  // (2,2048,2048)
  const float* ba   = (const float*)d_in[11];
  const float* Wr   = (const float*)d_in[12];  // (2,2048,8192)
  const float* br   = (const float*)d_in[13];
  const float* Wz   = (const float*)d_in[14];  // (2,2048,6144)
  const float* bz   = (const float*)d_in[15];
  const float* Wh   = (const float*)d_in[16];  // (2,2048,6144)
  const float* bh   = (const float*)d_in[17];

  const long long BD = (long long)BSZ * DMODEL;  // 32768
  float* w = (float*)d_ws;
  float* Sv    = w; w += BD;
  float* St    = w; w += BD;
  float* av    = w; w += BD;
  float* atb   = w; w += BD;
  float* fam   = w; w += BD;
  float* catv  = w; w += 2 * BD;
  float* catt  = w; w += 2 * BD;
  float* qh    = w; w += BD;
  float* Qk    = w; w += (long long)BSZ * NHEAD * DMODEL;   // (B,H,D)
  float* cK    = w; w += 256;
  float* attn  = w; w += (long long)BSZ * 1536 * NHEAD;     // (B,L,H) scores/probs
  float* ctxw  = w; w += (long long)BSZ * NHEAD * DMODEL;   // (B,H,D)
  float* ctx   = w; w += BD;
  float* vself = w; w += BD;
  float* t2v   = w; w += BD;
  float* tself = w; w += BD;
  float* v2t   = w; w += BD;
  float* cat   = w; w += (long long)BSZ * 4 * DMODEL;       // (B,8192)
  float* rb    = w; w += BD;
  float* zb    = w; w += BD;
  float* hb    = w; w += BD;

  const dim3 ew(128), ewt(256);
  const long long DD = (long long)DMODEL * DMODEL;

  // ---- precompute: means, a_v/a_t, Sv/St init ----
  mean_k<<<dim3(8, 16), 256, 0, stream>>>(Fa, 11, fam, DMODEL);
  mean_k<<<dim3(8, 16), 256, 0, stream>>>(Mv + 3LL * BSZ * 196 * DMODEL, 196, catv, 4096);
  mean_k<<<dim3(8, 16), 256, 0, stream>>>(Mv + 2LL * BSZ * 196 * DMODEL, 196, catv + 2048, 4096);
  mean_k<<<dim3(8, 16), 256, 0, stream>>>(Mt + 3LL * BSZ * 384 * DMODEL, 384, catt, 4096);
  mean_k<<<dim3(8, 16), 256, 0, stream>>>(Mt + 2LL * BSZ * 384 * DMODEL, 384, catt + 2048, 4096);

  la_at(stream, fam, DMODEL, 0, 0, 0, Wa, DMODEL, 0, ba, 0,
        av, DMODEL, 0, BSZ, DMODEL, DMODEL, 1.f, 0, 1);
  la_at(stream, fam, DMODEL, 0, 0, 0, Wa + DD, DMODEL, 0, ba + DMODEL, 0,
        atb, DMODEL, 0, BSZ, DMODEL, DMODEL, 1.f, 0, 1);
  la_at(stream, catv, 4096, 0, 0, 0, phiw, 4096, 0, phib, 0,
        Sv, DMODEL, 0, BSZ, DMODEL, 4096, 1.f, 1, 1);
  la_at(stream, catt, 4096, 0, 0, 0, phiw + 2048LL * 4096, 4096, 0, phib + DMODEL, 0,
        St, DMODEL, 0, BSZ, DMODEL, 4096, 1.f, 1, 1);

  // attention module table
  struct Cfg { int m; const float* q; const float* mem; int chunk; int L;
               const int* mask; float* dst; };
  const Cfg cfg[4] = {
    {0, Sv, Mv, 196,  784, nullptr, vself},
    {1, Sv, Mt, 384, 1536, tmk,     t2v  },
    {2, St, Mt, 384, 1536, tmk,     tself},
    {3, St, Mv, 196,  784, nullptr, v2t  },
  };

  for (int hop = 0; hop < 3; hop++) {
    // ---- 4 attention decodes (deferred K/V projection) ----
    for (int a = 0; a < 4; a++) {
      const Cfg& c = cfg[a];
      const float* Wq = Wqkv + (long long)c.m * 3 * DD;
      const float* Wk = Wq + DD;
      const float* Wv = Wk + DD;
      const float* bq = bqkv + (long long)c.m * 3 * DMODEL;
      const float* bk = bq + DMODEL;
      const float* bv = bk + DMODEL;

      // qh = q @ Wq^T + bq                  (16,2048)
      la_at(stream, c.q, DMODEL, 0, 0, 0, Wq, DMODEL, 0, bq, 0,
            qh, DMODEL, 0, BSZ, DMODEL, DMODEL, 1.f, 0, 1);
      // cK[b,h] = qh_h . bk_h
      ck_k<<<BSZ, 256, 0, stream>>>(qh, bk, cK);
      // Qk[b,h,:] = qh_h @ Wk_h             (batched over h)
      la_an(stream, qh, DMODEL, 1, HDIM, Wk, DMODEL, (long long)HDIM * DMODEL,
            0, 0, Qk, (long long)NHEAD * DMODEL, DMODEL, BSZ, DMODEL, HDIM, NHEAD);
      // scores[b,l,h] = (mem_row . Qk[b,h] + cK[b,h]) / 16   (batched over b)
      la_at(stream, c.mem, 0, 0, c.chunk, DMODEL,
            Qk, DMODEL, (long long)NHEAD * DMODEL, cK, NHEAD,
            attn, NHEAD, (long long)c.L * NHEAD,
            c.L, NHEAD, DMODEL, 1.0f / 16.0f, 0, BSZ);
      // softmax over l (with padding mask for text memories)
      softmax_k<<<BSZ * NHEAD, 256, 0, stream>>>(attn, c.L, c.mask, 384);
      // ctxw[b,h,:] = sum_l attn * mem_row  (batched over b)
      la_an(stream, attn, 1, NHEAD, (long long)c.L * NHEAD,
            c.mem, 0, 0, c.chunk, DMODEL,
            ctxw, DMODEL, (long long)NHEAD * DMODEL, NHEAD, DMODEL, c.L, BSZ);
      // ctx_h = ctxw_h @ Wv_h^T + bv_h      (batched over h)
      la_at(stream, ctxw, (long long)NHEAD * DMODEL, DMODEL, 0, 0,
            Wv, DMODEL, (long long)HDIM * DMODEL, bv, HDIM,
            ctx, DMODEL, HDIM, BSZ, HDIM, DMODEL, 1.f, 0, NHEAD);
      // dst = ctx @ Wo^T + bo
      la_at(stream, ctx, DMODEL, 0, 0, 0, Wo + (long long)c.m * DD, DMODEL, 0,
            bo + (long long)c.m * DMODEL, 0, c.dst, DMODEL, 0,
            BSZ, DMODEL, DMODEL, 1.f, 0, 1);
    }

    // ---- gated updates (v side then t side) ----
    for (int side = 0; side < 2; side++) {
      float* S          = side ? St : Sv;
      const float* self = side ? tself : vself;
      const float* cross= side ? v2t : t2v;
      const float* aa   = side ? atb : av;
      const float* Wrs  = Wr + (long long)side * DMODEL * 8192;
      const float* Wzs  = Wz + (long long)side * DMODEL * 6144;
      const float* Whs  = Wh + (long long)side * DMODEL * 6144;

      concat4_k<<<ew, ewt, 0, stream>>>(S, self, cross, aa, cat, 8192);
      la_at(stream, cat, 8192, 0, 0, 0, Wrs, 8192, 0, br + side * DMODEL, 0,
            rb, DMODEL, 0, BSZ, DMODEL, 8192, 1.f, 2, 1);
      la_at(stream, cat, 8192, 0, 0, 0, Wzs, 6144, 0, bz + side * DMODEL, 0,
            zb, DMODEL, 0, BSZ, DMODEL, 6144, 1.f, 2, 1);
      mulin_k<<<ew, ewt, 0, stream>>>(rb, S, cat, 8192);
      la_at(stream, cat, 8192, 0, 0, 0, Whs, 6144, 0, bh + side * DMODEL, 0,
            hb, DMODEL, 0, BSZ, DMODEL, 6144, 1.f, 1, 1);
      gru_k<<<ew, ewt, 0, stream>>>(zb, hb, S);
    }
  }

  final_k<<<ew, ewt, 0, stream>>>(Sv, St, (float*)d_out);
}